// Physics_4166118277764
// MI455X (gfx1250) — compile-verified
//
#include <hip/hip_runtime.h>
#include <cstdint>

#define TPB 256

// ---- CDNA5 async global<->LDS data mover (ASYNCcnt-tracked) ----
__device__ __forceinline__ void async_ld_b128(uint32_t lds, uint64_t gaddr) {
    asm volatile("global_load_async_to_lds_b128 %0, %1, off"
                 :: "v"(lds), "v"(gaddr) : "memory");
}
__device__ __forceinline__ void async_ld_b32(uint32_t lds, uint64_t gaddr) {
    asm volatile("global_load_async_to_lds_b32 %0, %1, off"
                 :: "v"(lds), "v"(gaddr) : "memory");
}
__device__ __forceinline__ void async_st_b128(uint64_t gaddr, uint32_t lds) {
    asm volatile("global_store_async_from_lds_b128 %0, %1, off"
                 :: "v"(gaddr), "v"(lds) : "memory");
}
__device__ __forceinline__ void async_st_b32(uint64_t gaddr, uint32_t lds) {
    asm volatile("global_store_async_from_lds_b32 %0, %1, off"
                 :: "v"(gaddr), "v"(lds) : "memory");
}
__device__ __forceinline__ void wait_async0() {
    asm volatile("s_wait_asynccnt 0" ::: "memory");
}

// Branchless exact Jacobi rotation on pivot (p,q); k is the remaining index.
// S' = J^T S J with t chosen to zero S_pq; V' = V J.
// If |S_pq| is negligible, t=0 -> c=1 (v_rsq_f32 exact on 1.0), s=0 -> exact no-op.
__device__ __forceinline__ void jrot(float& spp, float& sqq, float& spq,
                                     float& skp, float& skq,
                                     float& v0p, float& v0q,
                                     float& v1p, float& v1q,
                                     float& v2p, float& v2q) {
    float apq = spq;
    bool  ok  = fabsf(apq) > 1e-30f;
    float den = ok ? apq : 1.0f;
    float tau = (sqq - spp) * 0.5f * __builtin_amdgcn_rcpf(den);
    // t = sgn(tau) / (|tau| + sqrt(1+tau^2));  tau->inf gives t->0 gracefully
    float t0  = __builtin_amdgcn_rcpf(fabsf(tau) + sqrtf(fmaf(tau, tau, 1.0f)));
    float t   = ok ? copysignf(t0, tau) : 0.0f;
    float c   = __builtin_amdgcn_rsqf(fmaf(t, t, 1.0f));
    float s   = t * c;
    float tapq = t * apq;
    spp -= tapq;
    sqq += tapq;
    spq  = 0.0f;                    // |spq| <= 1e-30 when rotation skipped: harmless
    float kp = skp, kq = skq;
    skp = fmaf(c, kp, -s * kq);
    skq = fmaf(s, kp,  c * kq);
    float x;
    x = v0p; v0p = fmaf(c, x, -s * v0q); v0q = fmaf(s, x, c * v0q);
    x = v1p; v1p = fmaf(c, x, -s * v1q); v1q = fmaf(s, x, c * v1q);
    x = v2p; v2p = fmaf(c, x, -s * v2q); v2q = fmaf(s, x, c * v2q);
}

__global__ __launch_bounds__(TPB)
void plastic_svd_kernel(const float* __restrict__ F,
                        const float* __restrict__ p_yml,
                        const float* __restrict__ p_nu,
                        const float* __restrict__ p_ys,
                        float* __restrict__ out, int N) {
    __shared__ __align__(16) float smem[TPB * 9];

    const int nb          = min(TPB, N - (int)blockIdx.x * TPB); // matrices in this block
    const uint32_t bytes  = (uint32_t)nb * 36u;
    const uint32_t n128   = bytes >> 4;          // whole 16B chunks
    const uint32_t ntail  = (bytes & 15u) >> 2;  // leftover dwords
    const uint64_t gin    = (uint64_t)(uintptr_t)(F + (size_t)blockIdx.x * TPB * 9);
    const uint32_t lds0   = (uint32_t)(uintptr_t)&smem[0]; // low 32 bits = LDS offset

    // ---- async stage: global -> LDS (128-bit aligned transactions) ----
    for (uint32_t i = threadIdx.x; i < n128; i += TPB)
        async_ld_b128(lds0 + i * 16u, gin + i * 16u);
    if (threadIdx.x < ntail) {
        uint32_t off = n128 * 16u + threadIdx.x * 4u;
        async_ld_b32(lds0 + off, gin + off);
    }
    wait_async0();
    __syncthreads();

    // uniform scalar parameters (scalar-cache loads); precise exp once per thread
    const float yml = p_yml[0];
    const float nu  = p_nu[0];
    const float ys  = p_ys[0];
    const float mu   = expf(yml) / (2.0f * (1.0f + nu));
    const float kfac = ys / (2.0f * mu);

    const int t    = (int)threadIdx.x;
    const int gidx = (int)blockIdx.x * TPB + t;
    if (gidx < N) {
        const float* m = &smem[t * 9];
        float a00 = m[0], a01 = m[1], a02 = m[2];
        float a10 = m[3], a11 = m[4], a12 = m[5];
        float a20 = m[6], a21 = m[7], a22 = m[8];

        // S = A^T A (symmetric, PSD)
        float s00 = a00*a00 + a10*a10 + a20*a20;
        float s01 = a00*a01 + a10*a11 + a20*a21;
        float s02 = a00*a02 + a10*a12 + a20*a22;
        float s11 = a01*a01 + a11*a11 + a21*a21;
        float s12 = a01*a02 + a11*a12 + a21*a22;
        float s22 = a02*a02 + a12*a12 + a22*a22;

        // V = I; cyclic Jacobi, 4 sweeps (quadratic convergence, f32 precision)
        float v00 = 1.f, v01 = 0.f, v02 = 0.f;
        float v10 = 0.f, v11 = 1.f, v12 = 0.f;
        float v20 = 0.f, v21 = 0.f, v22 = 1.f;
        #pragma unroll
        for (int it = 0; it < 4; ++it) {
            jrot(s00, s11, s01, s02, s12, v00, v01, v10, v11, v20, v21); // (0,1), k=2
            jrot(s00, s22, s02, s01, s12, v00, v02, v10, v12, v20, v22); // (0,2), k=1
            jrot(s11, s22, s12, s01, s02, v01, v02, v11, v12, v21, v22); // (1,2), k=0
        }

        // B = A V ; sigma_i = ||column i of B|| ; U = B diag(1/sigma) implicitly
        float b00 = a00*v00 + a01*v10 + a02*v20;
        float b01 = a00*v01 + a01*v11 + a02*v21;
        float b02 = a00*v02 + a01*v12 + a02*v22;
        float b10 = a10*v00 + a11*v10 + a12*v20;
        float b11 = a10*v01 + a11*v11 + a12*v21;
        float b12 = a10*v02 + a11*v12 + a12*v22;
        float b20 = a20*v00 + a21*v10 + a22*v20;
        float b21 = a20*v01 + a21*v11 + a22*v21;
        float b22 = a20*v02 + a21*v12 + a22*v22;

        float sg0 = sqrtf(b00*b00 + b10*b10 + b20*b20);
        float sg1 = sqrtf(b01*b01 + b11*b11 + b21*b21);
        float sg2 = sqrtf(b02*b02 + b12*b12 + b22*b22);

        // von-Mises return mapping on Hencky strain of singular values
        float sc0 = fmaxf(sg0, 0.01f);
        float sc1 = fmaxf(sg1, 0.01f);
        float sc2 = fmaxf(sg2, 0.01f);
        float e0 = __logf(sc0), e1 = __logf(sc1), e2 = __logf(sc2);
        float mtr = (e0 + e1 + e2) * (1.0f / 3.0f);
        float d0 = e0 - mtr, d1 = e1 - mtr, d2 = e2 - mtr;
        float nrm = sqrtf(d0*d0 + d1*d1 + d2*d2) + 1e-5f;
        float dg  = nrm - kfac;
        // branchless: compute fast exp unconditionally, select exact sc when elastic
        float fr  = dg * __builtin_amdgcn_rcpf(nrm);
        bool  yld = dg > 0.0f;
        float sn0 = yld ? __expf(fmaf(-fr, d0, e0)) : sc0;
        float sn1 = yld ? __expf(fmaf(-fr, d1, e1)) : sc1;
        float sn2 = yld ? __expf(fmaf(-fr, d2, e2)) : sc2;

        // F_corrected = U diag(sn) V^T = B diag(sn/sigma) V^T
        float r0 = sn0 * __builtin_amdgcn_rcpf(fmaxf(sg0, 1e-20f));
        float r1 = sn1 * __builtin_amdgcn_rcpf(fmaxf(sg1, 1e-20f));
        float r2 = sn2 * __builtin_amdgcn_rcpf(fmaxf(sg2, 1e-20f));
        b00 *= r0; b10 *= r0; b20 *= r0;
        b01 *= r1; b11 *= r1; b21 *= r1;
        b02 *= r2; b12 *= r2; b22 *= r2;

        float* o = &smem[t * 9];
        o[0] = b00*v00 + b01*v01 + b02*v02;
        o[1] = b00*v10 + b01*v11 + b02*v12;
        o[2] = b00*v20 + b01*v21 + b02*v22;
        o[3] = b10*v00 + b11*v01 + b12*v02;
        o[4] = b10*v10 + b11*v11 + b12*v12;
        o[5] = b10*v20 + b11*v21 + b12*v22;
        o[6] = b20*v00 + b21*v01 + b22*v02;
        o[7] = b20*v10 + b21*v11 + b22*v12;
        o[8] = b20*v20 + b21*v21 + b22*v22;
    }
    __syncthreads(); // waits DScnt: LDS results visible before async store reads them

    // ---- async stage: LDS -> global ----
    const uint64_t gout = (uint64_t)(uintptr_t)(out + (size_t)blockIdx.x * TPB * 9);
    for (uint32_t i = threadIdx.x; i < n128; i += TPB)
        async_st_b128(gout + i * 16u, lds0 + i * 16u);
    if (threadIdx.x < ntail) {
        uint32_t off = n128 * 16u + threadIdx.x * 4u;
        async_st_b32(gout + off, lds0 + off);
    }
    wait_async0();
}

extern "C" void kernel_launch(void* const* d_in, const int* in_sizes, int n_in,
                              void* d_out, int out_size, void* d_ws, size_t ws_size,
                              hipStream_t stream) {
    const float* F     = (const float*)d_in[0];
    const float* p_yml = (const float*)d_in[1];
    const float* p_nu  = (const float*)d_in[2];
    const float* p_ys  = (const float*)d_in[3];
    float* out = (float*)d_out;
    int N = in_sizes[0] / 9;
    int blocks = (N + TPB - 1) / TPB;
    hipLaunchKernelGGL(plastic_svd_kernel, dim3(blocks), dim3(TPB), 0, stream,
                       F, p_yml, p_nu, p_ys, out, N);
}